// BaseViewTransform_45432164057423
// MI455X (gfx1250) — compile-verified
//
#include <hip/hip_runtime.h>

typedef __attribute__((ext_vector_type(2))) float v2f;
typedef __attribute__((ext_vector_type(8))) float v8f;

#define C_CH   80
#define GNX    360
#define GNY    360
#define XY     (GNX * GNY)          /* 129600 */
#define PTS_PER_BLOCK 128
#define NTHREADS      256

// ---------------------------------------------------------------------------
// Kernel 1: per-batch combine = rots @ inv(intrins), packed as rows [c0 c1 c2 t]
// ---------------------------------------------------------------------------
__global__ void bev_prep(const float* __restrict__ rots,
                         const float* __restrict__ trans,
                         const float* __restrict__ intr,
                         float* __restrict__ ws, int B) {
    int b = threadIdx.x;
    if (b >= B) return;
    const float* K = intr + b * 9;
    float a = K[0], bb = K[1], c = K[2];
    float d = K[3], e  = K[4], f = K[5];
    float g = K[6], h  = K[7], i = K[8];
    // adjugate / det
    float A0 = e*i - f*h, A1 = c*h - bb*i, A2 = bb*f - c*e;
    float A3 = f*g - d*i, A4 = a*i - c*g,  A5 = c*d - a*f;
    float A6 = d*h - e*g, A7 = bb*g - a*h, A8 = a*e - bb*d;
    float rdet = 1.0f / (a*A0 + bb*A3 + c*A6);
    float inv[9] = {A0*rdet, A1*rdet, A2*rdet,
                    A3*rdet, A4*rdet, A5*rdet,
                    A6*rdet, A7*rdet, A8*rdet};
    const float* R = rots + b * 9;
    for (int r = 0; r < 3; ++r) {
        for (int cc = 0; cc < 3; ++cc) {
            ws[b*12 + r*4 + cc] = R[r*3+0]*inv[0*3+cc]
                                + R[r*3+1]*inv[1*3+cc]
                                + R[r*3+2]*inv[2*3+cc];
        }
        ws[b*12 + r*4 + 3] = trans[b*3 + r];
    }
}

// ---------------------------------------------------------------------------
// Kernel 2: zero the output (harness poisons it with 0xAA before timing)
// ---------------------------------------------------------------------------
__global__ void bev_zero(float* __restrict__ out, long n) {
    long n4 = n >> 2;
    float4* out4 = (float4*)out;
    float4 z; z.x = z.y = z.z = z.w = 0.0f;
    long stride = (long)gridDim.x * blockDim.x;
    for (long idx = (long)blockIdx.x * blockDim.x + threadIdx.x; idx < n4; idx += stride)
        out4[idx] = z;
    long tail = n - (n4 << 2);
    if (blockIdx.x == 0 && threadIdx.x < tail)
        out[(n4 << 2) + threadIdx.x] = 0.0f;
}

// ---------------------------------------------------------------------------
// Kernel 3: geometry via V_WMMA_F32_16X16X4_F32 + atomic scatter of features
// ---------------------------------------------------------------------------
__global__ __launch_bounds__(NTHREADS)
void bev_scatter(const float* __restrict__ x,
                 const float* __restrict__ frustum,
                 const float* __restrict__ ws,
                 float* __restrict__ out,
                 long Np, long PB) {
    __shared__ int smBase[PTS_PER_BLOCK];

    const int  tid        = threadIdx.x;
    const long block_base = (long)blockIdx.x * PTS_PER_BLOCK;
    const long bbc        = (block_base < Np) ? block_base : (Np - 1);
    const int  b          = (int)(bbc / PB);      // batch uniform per block (PB % 128 == 0)

    // ---- Phase 1: per-wave WMMA computes geom = [combine|trans] @ (u*d, v*d, d, 1)
    {
        const int  lane = tid & 31;
        const int  wave = tid >> 5;
        const int  p16  = lane & 15;
        const int  hi   = lane >> 4;              // 0: K0/K1 half, 1: K2/K3 half
        const long pp   = block_base + (long)wave * 16 + p16;
        const long ppc  = (pp < Np) ? pp : (Np - 1);
        const long fidx = ppc - (long)b * PB;     // index into frustum (D*H*W, 3)

        const float* fr = frustum + fidx * 3;
        float u  = fr[0];
        float vv = fr[1];
        float dd = fr[2];

        // B matrix (4x16): rows K0=u*d, K1=v*d, K2=d, K3=1; VGPR0={K0|K2}, VGPR1={K1|K3}
        v2f Bop;
        Bop.x = hi ? dd   : u  * dd;
        Bop.y = hi ? 1.0f : vv * dd;

        // A matrix (16x4): rows 0..2 = [combine row | trans], rows 3..15 = 0
        float a0 = 0.0f, a1 = 0.0f;
        if (p16 < 3) {
            const float* Mb = ws + b * 12 + p16 * 4;
            a0 = hi ? Mb[2] : Mb[0];
            a1 = hi ? Mb[3] : Mb[1];
        }
        v2f Aop; Aop.x = a0; Aop.y = a1;

        v8f Cz = {0.f, 0.f, 0.f, 0.f, 0.f, 0.f, 0.f, 0.f};
        // D(16x16) = A(16x4) x B(4x16); rows 0..2 = geom x/y/z for 16 points
        v8f D = __builtin_amdgcn_wmma_f32_16x16x4_f32(
            false, Aop, false, Bop, (short)0, Cz, false, false);

        if (lane < 16) {
            float gx = D[0];   // M=0, N=lane -> x of point p16
            float gy = D[1];   // M=1
            float gz = D[2];   // M=2
            // replicate reference exactly: (geom - BX_LO)/DX, cast (trunc), then bounds
            int cx = (int)((gx + 54.0f) / 0.3f);
            int cy = (int)((gy + 54.0f) / 0.3f);
            int cz = (int)((gz + 10.0f) / 20.0f);
            int base = -1;
            if (pp < Np && cx >= 0 && cx < GNX && cy >= 0 && cy < GNY && cz == 0)
                base = b * (C_CH * XY) + cx * GNY + cy;
            smBase[wave * 16 + p16] = base;
        }
    }
    __syncthreads();

    // ---- Phase 2: coalesced float4 feature loads + native f32 atomics into L2
    const float4* x4   = (const float4*)x;
    const long    blk4 = block_base * (C_CH / 4);    // float4 offset of this block
    #pragma unroll
    for (int it = 0; it < (PTS_PER_BLOCK * (C_CH / 4)) / NTHREADS; ++it) {  // 10 iters
        int e4   = tid + it * NTHREADS;
        int p    = e4 / (C_CH / 4);                  // /20
        int c4   = e4 - p * (C_CH / 4);
        int base = smBase[p];
        if (base >= 0) {
            float4 v = x4[blk4 + e4];                // global_load_b128, fully coalesced
            unsigned long long a0 = (unsigned long long)(out + (size_t)base + (size_t)(c4 * 4) * XY);
            unsigned long long a1 = a0 + (unsigned long long)XY * 4ull;
            unsigned long long a2 = a1 + (unsigned long long)XY * 4ull;
            unsigned long long a3 = a2 + (unsigned long long)XY * 4ull;
            // native CDNA5 L2 atomics (avoid HIP's CAS-loop fallback for float)
            asm volatile("global_atomic_add_f32 %0, %1, off" :: "v"(a0), "v"(v.x) : "memory");
            asm volatile("global_atomic_add_f32 %0, %1, off" :: "v"(a1), "v"(v.y) : "memory");
            asm volatile("global_atomic_add_f32 %0, %1, off" :: "v"(a2), "v"(v.z) : "memory");
            asm volatile("global_atomic_add_f32 %0, %1, off" :: "v"(a3), "v"(v.w) : "memory");
        }
    }
}

// ---------------------------------------------------------------------------
extern "C" void kernel_launch(void* const* d_in, const int* in_sizes, int n_in,
                              void* d_out, int out_size, void* d_ws, size_t ws_size,
                              hipStream_t stream) {
    const float* x       = (const float*)d_in[0];   // (B, D, fH, fW, C) f32
    const float* rots    = (const float*)d_in[1];   // (B, 3, 3)
    const float* trans   = (const float*)d_in[2];   // (B, 3)
    const float* intr    = (const float*)d_in[3];   // (B, 3, 3)
    const float* frustum = (const float*)d_in[4];   // (D, fH, fW, 3)
    float* out = (float*)d_out;
    float* wsF = (float*)d_ws;

    const int  B  = in_sizes[1] / 9;
    const long Np = (long)in_sizes[0] / C_CH;       // total points
    const long PB = Np / B;                         // points per batch (divisible by 128)

    bev_prep<<<1, 32, 0, stream>>>(rots, trans, intr, wsF, B);
    bev_zero<<<2048, NTHREADS, 0, stream>>>(out, (long)out_size);
    const int nblk = (int)((Np + PTS_PER_BLOCK - 1) / PTS_PER_BLOCK);
    bev_scatter<<<nblk, NTHREADS, 0, stream>>>(x, frustum, wsF, out, Np, PB);
}